// TFNEigenNet_59751585022583
// MI455X (gfx1250) — compile-verified
//
#include <hip/hip_runtime.h>
#include <math.h>

#define NPTS 512
#define CH 32
#define EPSF 1e-8f
#define GAMMA 1.1428571428571428f   // 1 / (3.5/4)

typedef __attribute__((ext_vector_type(2))) float v2f;
typedef __attribute__((ext_vector_type(8))) float v8f;

struct RadialP {
  const float* w1;  // [4][4]
  const float* b1;  // [4]
  const float* w2;  // [32][4]
  const float* b2;  // [32]
};

__device__ __forceinline__ float ssp(float x) {
  // log(exp(x)+1) - log(2), numerically stable
  return fmaxf(x, 0.0f) + log1pf(expf(-fabsf(x))) - 0.69314718055994531f;
}

__device__ __forceinline__ void rbf_calc(float dij, float rbf[4]) {
  const float c0 = 0.0f, c1 = 1.1666666666666667f, c2 = 2.3333333333333335f, c3 = 3.5f;
  float d;
  d = dij - c0; rbf[0] = expf(-GAMMA * d * d);
  d = dij - c1; rbf[1] = expf(-GAMMA * d * d);
  d = dij - c2; rbf[2] = expf(-GAMMA * d * d);
  d = dij - c3; rbf[3] = expf(-GAMMA * d * d);
}

// ---------------------------------------------------------------------------
// Layer 1: b-independent input feature => radial 2nd layer factors out of the
// N^2 loop. One wave per point 'a'. Produces t0[512][32], t1[512][32][3].
// ---------------------------------------------------------------------------
__global__ __launch_bounds__(256) void tfn_layer1(
    const float* __restrict__ x, const float* __restrict__ ew,
    RadialP p0, RadialP p1,
    const float* __restrict__ siw0, const float* __restrict__ sib0,
    const float* __restrict__ siw1,
    const float* __restrict__ nlb0, const float* __restrict__ nlb1,
    float* __restrict__ t0out, float* __restrict__ t1out)
{
  __shared__ float lcat[8][CH][4];  // per-wave: cat0 + cat1[3] per channel
  const int w = threadIdx.x >> 5;
  const int lane = threadIdx.x & 31;
  const int a = blockIdx.x * 8 + w;

  const float xa0 = x[a * 3 + 0], xa1 = x[a * 3 + 1], xa2 = x[a * 3 + 2];

  float s_h0[4] = {0.f, 0.f, 0.f, 0.f};
  float s_hu[4][3] = {};
  float s_u[3] = {0.f, 0.f, 0.f};

  for (int b = lane; b < NPTS; b += 32) {
    float r0 = xa0 - x[b * 3 + 0];
    float r1 = xa1 - x[b * 3 + 1];
    float r2 = xa2 - x[b * 3 + 2];
    float d2 = r0 * r0 + r1 * r1 + r2 * r2;
    float dij = sqrtf(fmaxf(d2, EPSF));
    float inv = 1.0f / dij;
    bool msk = d2 < EPSF;
    float u0 = msk ? 0.f : r0 * inv;
    float u1 = msk ? 0.f : r1 * inv;
    float u2 = msk ? 0.f : r2 * inv;
    float rbf[4];
    rbf_calc(dij, rbf);
    #pragma unroll
    for (int h = 0; h < 4; ++h) {
      float h0 = p0.b1[h], h1 = p1.b1[h];
      #pragma unroll
      for (int k = 0; k < 4; ++k) {
        h0 = fmaf(p0.w1[h * 4 + k], rbf[k], h0);
        h1 = fmaf(p1.w1[h * 4 + k], rbf[k], h1);
      }
      h0 = fmaxf(h0, 0.f);
      h1 = fmaxf(h1, 0.f);
      s_h0[h] += h0;
      s_hu[h][0] = fmaf(h1, u0, s_hu[h][0]);
      s_hu[h][1] = fmaf(h1, u1, s_hu[h][1]);
      s_hu[h][2] = fmaf(h1, u2, s_hu[h][2]);
    }
    s_u[0] += u0; s_u[1] += u1; s_u[2] += u2;
  }

  // wave-wide reduction of the 19 sums
  #pragma unroll
  for (int off = 16; off >= 1; off >>= 1) {
    #pragma unroll
    for (int h = 0; h < 4; ++h) {
      s_h0[h] += __shfl_xor(s_h0[h], off, 32);
      #pragma unroll
      for (int i = 0; i < 3; ++i) s_hu[h][i] += __shfl_xor(s_hu[h][i], off, 32);
    }
    #pragma unroll
    for (int i = 0; i < 3; ++i) s_u[i] += __shfl_xor(s_u[i], off, 32);
  }

  // per-channel conv outputs (lane == channel f)
  {
    const int f = lane;
    float c0 = 512.0f * p0.b2[f];
    float c1[3];
    #pragma unroll
    for (int i = 0; i < 3; ++i) c1[i] = p1.b2[f] * s_u[i];
    #pragma unroll
    for (int h = 0; h < 4; ++h) {
      c0 = fmaf(p0.w2[f * 4 + h], s_h0[h], c0);
      #pragma unroll
      for (int i = 0; i < 3; ++i) c1[i] = fmaf(p1.w2[f * 4 + h], s_hu[h][i], c1[i]);
    }
    float e = ew[f];
    lcat[w][f][0] = e * c0;
    lcat[w][f][1] = e * c1[0];
    lcat[w][f][2] = e * c1[1];
    lcat[w][f][3] = e * c1[2];
  }
  asm volatile("s_wait_dscnt 0" ::: "memory");

  // self-interaction + equivariant nonlinearity (lane == output channel g)
  {
    const int g = lane;
    float t0v = sib0[g] + nlb0[g];
    float tv[3] = {0.f, 0.f, 0.f};
    #pragma unroll 4
    for (int f = 0; f < CH; ++f) {
      t0v = fmaf(siw0[g * CH + f], lcat[w][f][0], t0v);
      float wv = siw1[g * CH + f];
      tv[0] = fmaf(wv, lcat[w][f][1], tv[0]);
      tv[1] = fmaf(wv, lcat[w][f][2], tv[1]);
      tv[2] = fmaf(wv, lcat[w][f][3], tv[2]);
    }
    t0v = ssp(t0v);
    float n2 = tv[0] * tv[0] + tv[1] * tv[1] + tv[2] * tv[2];
    float n1 = sqrtf(fmaxf(n2, EPSF));
    float sc = ssp(n1 + nlb1[g]) / n1;
    t0out[a * CH + g] = t0v;
    float* o = t1out + (size_t)(a * CH + g) * 3;
    o[0] = tv[0] * sc; o[1] = tv[1] * sc; o[2] = tv[2] * sc;
  }
}

// ---------------------------------------------------------------------------
// Layer 2: per 16-pair chunk, the radial 2nd layer R = h @ w2^T + b2 is a
// K=4 GEMM -> V_WMMA_F32_16X16X4_F32 (A: 16 pairs x 4 hidden, B: 4 x 16
// channels, C preloaded with bias). 3 live radials x 2 channel halves = 6
// WMMAs per chunk. Conv accumulation + SI + nonlinearity fused.
// ---------------------------------------------------------------------------
__global__ __launch_bounds__(256) void tfn_layer2(
    const float* __restrict__ x,
    RadialP pb, RadialP pc, RadialP pe,
    const float* __restrict__ siw1, const float* __restrict__ nlb1,
    const float* __restrict__ t0, const float* __restrict__ t1,
    float* __restrict__ out)
{
  __shared__ float lunit[8][16][3];
  __shared__ float lcat[8][96][3];
  const int w = threadIdx.x >> 5;
  const int lane = threadIdx.x & 31;
  const int a = blockIdx.x * 8 + w;
  const int m = lane & 15;          // pair row within chunk (A layout: M = lane%16)
  const int kb = (lane >> 4) * 2;   // hidden K base held by this lane (0 or 2)
  const int rh = lane >> 4;         // D-tile row half

  const float xa0 = x[a * 3 + 0], xa1 = x[a * 3 + 1], xa2 = x[a * 3 + 2];

  // B matrices (w2^T) and bias, per radial per channel-half; loop-invariant
  v2f Bb[2], Bc[2], Be[2];
  float bb2[2], bc2[2], be2[2];
  #pragma unroll
  for (int ch = 0; ch < 2; ++ch) {
    int f = ch * 16 + m;
    Bb[ch].x = pb.w2[f * 4 + kb]; Bb[ch].y = pb.w2[f * 4 + kb + 1];
    Bc[ch].x = pc.w2[f * 4 + kb]; Bc[ch].y = pc.w2[f * 4 + kb + 1];
    Be[ch].x = pe.w2[f * 4 + kb]; Be[ch].y = pe.w2[f * 4 + kb + 1];
    bb2[ch] = pb.b2[f]; bc2[ch] = pc.b2[f]; be2[ch] = pe.b2[f];
  }

  float acc0[2][3] = {};  // 0x1->1  (R_b * unit_i * t0)
  float accL[2][3] = {};  // Lx0     (R_c * t1_i)
  float accX[2][3] = {};  // cross   (R_e * (unit x t1)_i)

  for (int c = 0; c < NPTS / 16; ++c) {
    const int bp = c * 16 + m;
    float r0 = xa0 - x[bp * 3 + 0];
    float r1 = xa1 - x[bp * 3 + 1];
    float r2 = xa2 - x[bp * 3 + 2];
    float d2 = r0 * r0 + r1 * r1 + r2 * r2;
    float dij = sqrtf(fmaxf(d2, EPSF));
    float inv = 1.0f / dij;
    bool msk = d2 < EPSF;
    float u0 = msk ? 0.f : r0 * inv;
    float u1 = msk ? 0.f : r1 * inv;
    float u2 = msk ? 0.f : r2 * inv;
    float rbf[4];
    rbf_calc(dij, rbf);
    if (lane < 16) {
      lunit[w][m][0] = u0; lunit[w][m][1] = u1; lunit[w][m][2] = u2;
    }

    // A matrices: this lane's two hidden units (K = kb, kb+1) for pair m
    v2f Ab, Ac, Ae;
    {
      float hb0 = pb.b1[kb], hb1 = pb.b1[kb + 1];
      float hc0 = pc.b1[kb], hc1 = pc.b1[kb + 1];
      float he0 = pe.b1[kb], he1 = pe.b1[kb + 1];
      #pragma unroll
      for (int k = 0; k < 4; ++k) {
        hb0 = fmaf(pb.w1[kb * 4 + k], rbf[k], hb0);
        hb1 = fmaf(pb.w1[(kb + 1) * 4 + k], rbf[k], hb1);
        hc0 = fmaf(pc.w1[kb * 4 + k], rbf[k], hc0);
        hc1 = fmaf(pc.w1[(kb + 1) * 4 + k], rbf[k], hc1);
        he0 = fmaf(pe.w1[kb * 4 + k], rbf[k], he0);
        he1 = fmaf(pe.w1[(kb + 1) * 4 + k], rbf[k], he1);
      }
      Ab.x = fmaxf(hb0, 0.f); Ab.y = fmaxf(hb1, 0.f);
      Ac.x = fmaxf(hc0, 0.f); Ac.y = fmaxf(hc1, 0.f);
      Ae.x = fmaxf(he0, 0.f); Ae.y = fmaxf(he1, 0.f);
    }
    asm volatile("s_wait_dscnt 0" ::: "memory");  // lunit visible within wave

    #pragma unroll
    for (int ch = 0; ch < 2; ++ch) {
      v8f Cb, Cc, Ce;
      #pragma unroll
      for (int j = 0; j < 8; ++j) { Cb[j] = bb2[ch]; Cc[j] = bc2[ch]; Ce[j] = be2[ch]; }
      // R tiles: 16 pairs x 16 channels, f32, K=4
      v8f Db = __builtin_amdgcn_wmma_f32_16x16x4_f32(false, Ab, false, Bb[ch], (short)0, Cb, false, false);
      v8f Dc = __builtin_amdgcn_wmma_f32_16x16x4_f32(false, Ac, false, Bc[ch], (short)0, Cc, false, false);
      v8f De = __builtin_amdgcn_wmma_f32_16x16x4_f32(false, Ae, false, Be[ch], (short)0, Ce, false, false);

      const int f = ch * 16 + m;
      #pragma unroll
      for (int v = 0; v < 8; ++v) {
        const int row = v + rh * 8;           // D layout: lanes>=16 hold M=8..15
        const int bi = c * 16 + row;
        float tt0 = t0[bi * CH + f];
        const float* tp = t1 + (size_t)(bi * CH + f) * 3;
        float t1x = tp[0], t1y = tp[1], t1z = tp[2];
        float uu0 = lunit[w][row][0];
        float uu1 = lunit[w][row][1];
        float uu2 = lunit[w][row][2];
        float Rb = Db[v], Rc = Dc[v], Re = De[v];
        float rt = Rb * tt0;
        acc0[ch][0] = fmaf(rt, uu0, acc0[ch][0]);
        acc0[ch][1] = fmaf(rt, uu1, acc0[ch][1]);
        acc0[ch][2] = fmaf(rt, uu2, acc0[ch][2]);
        accL[ch][0] = fmaf(Rc, t1x, accL[ch][0]);
        accL[ch][1] = fmaf(Rc, t1y, accL[ch][1]);
        accL[ch][2] = fmaf(Rc, t1z, accL[ch][2]);
        float cx0 = uu1 * t1z - uu2 * t1y;
        float cx1 = uu2 * t1x - uu0 * t1z;
        float cx2 = uu0 * t1y - uu1 * t1x;
        accX[ch][0] = fmaf(Re, cx0, accX[ch][0]);
        accX[ch][1] = fmaf(Re, cx1, accX[ch][1]);
        accX[ch][2] = fmaf(Re, cx2, accX[ch][2]);
      }
    }
  }

  // combine row halves (lane n and n+16 hold the two 8-row partials of channel n)
  #pragma unroll
  for (int ch = 0; ch < 2; ++ch)
    #pragma unroll
    for (int i = 0; i < 3; ++i) {
      acc0[ch][i] += __shfl_xor(acc0[ch][i], 16, 32);
      accL[ch][i] += __shfl_xor(accL[ch][i], 16, 32);
      accX[ch][i] += __shfl_xor(accX[ch][i], 16, 32);
    }

  if (lane < 16) {
    #pragma unroll
    for (int ch = 0; ch < 2; ++ch) {
      int f = ch * 16 + lane;
      #pragma unroll
      for (int i = 0; i < 3; ++i) {
        lcat[w][f][i] = acc0[ch][i];        // channels 0..31: 0x1->1
        lcat[w][32 + f][i] = accL[ch][i];   // channels 32..63: Lx0
        lcat[w][64 + f][i] = accX[ch][i];   // channels 64..95: cross
      }
    }
  }
  asm volatile("s_wait_dscnt 0" ::: "memory");

  // self-interaction (l=1, no bias) + norm nonlinearity; lane == channel g
  {
    const int g = lane;
    float tv[3] = {0.f, 0.f, 0.f};
    #pragma unroll 4
    for (int f = 0; f < 96; ++f) {
      float wv = siw1[g * 96 + f];
      tv[0] = fmaf(wv, lcat[w][f][0], tv[0]);
      tv[1] = fmaf(wv, lcat[w][f][1], tv[1]);
      tv[2] = fmaf(wv, lcat[w][f][2], tv[2]);
    }
    float n2 = tv[0] * tv[0] + tv[1] * tv[1] + tv[2] * tv[2];
    float n1 = sqrtf(fmaxf(n2, EPSF));
    float sc = ssp(n1 + nlb1[g]) / n1;
    float* o = out + (size_t)(a * CH + g) * 3;
    o[0] = tv[0] * sc; o[1] = tv[1] * sc; o[2] = tv[2] * sc;
  }
}

// ---------------------------------------------------------------------------
extern "C" void kernel_launch(void* const* d_in, const int* in_sizes, int n_in,
                              void* d_out, int out_size, void* d_ws, size_t ws_size,
                              hipStream_t stream)
{
  auto in = [&](int i) { return (const float*)d_in[i]; };

  const float *x, *ew;
  RadialP p0, p1, pb, pc, pe;
  const float *si0w0, *si0b0, *si0w1, *nl0b0, *nl0b1, *si1w1, *nl1b1;

  if (n_in > 0 && in_sizes[0] == 3 * NPTS) {
    // insertion-order flatten: x, embed_w, then per-layer {conv(w1,b1,w2,b2)..., si(w0,b0,w1), nl(b0,b1)}
    x = in(0); ew = in(1);
    auto rad = [&](int base) { RadialP r; r.w1 = in(base); r.b1 = in(base + 1); r.w2 = in(base + 2); r.b2 = in(base + 3); return r; };
    p0 = rad(2); p1 = rad(6);
    si0w0 = in(10); si0b0 = in(11); si0w1 = in(12); nl0b0 = in(13); nl0b1 = in(14);
    pb = rad(19); pc = rad(23); pe = rad(31);   // layer-1 conv indices 1, 2, 4
    si1w1 = in(37); nl1b1 = in(39);
  } else {
    // jax-pytree sorted-key flatten: params first ('params' < 'x'), dict keys alphabetical
    // radial dicts: (b1,b2,w1,w2); layer dicts: conv, nl(b0,b1), si(b0,w0,w1)
    ew = in(0);
    auto rad = [&](int base) { RadialP r; r.b1 = in(base); r.b2 = in(base + 1); r.w1 = in(base + 2); r.w2 = in(base + 3); return r; };
    p0 = rad(1); p1 = rad(5);
    nl0b0 = in(9); nl0b1 = in(10); si0b0 = in(11); si0w0 = in(12); si0w1 = in(13);
    pb = rad(18); pc = rad(22); pe = rad(30);
    nl1b1 = in(35); si1w1 = in(38);
    x = in(39);
  }

  float* t0 = (float*)d_ws;           // [512][32]
  float* t1 = t0 + NPTS * CH;         // [512][32][3]

  tfn_layer1<<<NPTS / 8, 256, 0, stream>>>(x, ew, p0, p1, si0w0, si0b0, si0w1,
                                           nl0b0, nl0b1, t0, t1);
  tfn_layer2<<<NPTS / 8, 256, 0, stream>>>(x, pb, pc, pe, si1w1, nl1b1, t0, t1,
                                           (float*)d_out);
}